// EuclideanCodebook_90709709291559
// MI455X (gfx1250) — compile-verified
//
#include <hip/hip_runtime.h>
#include <hip/hip_bf16.h>
#include <math.h>

// ---------------------------------------------------------------------------
// EuclideanCodebook (VQ-VAE) step for MI455X / gfx1250 (wave32, WMMA)
//   B=16, S=2048, D=512, K=2048, N = B*S = 32768
// dist = 2*(x @ E^T) - |x|^2 - |e|^2 via V_WMMA_F32_16X16X4_F32,
// staged through LDS with GLOBAL_LOAD_ASYNC_TO_LDS_B128 (double-buffered),
// argmax fused into the GEMM epilogue via packed u64 atomicMax.
// ---------------------------------------------------------------------------

#define DECAY   0.99f
#define EPSILON 1e-5f

typedef float v2f __attribute__((ext_vector_type(2)));
typedef float v8f __attribute__((ext_vector_type(8)));
typedef int   v4i __attribute__((vector_size(4 * sizeof(int))));   // 16B block

// ---------------- tile config for the WMMA distance GEMM -------------------
#define TM    128    // rows of x per block
#define TN    64     // codebook columns per block
#define DC    32     // D-chunk staged in LDS per iteration
#define LDSS  36     // padded LDS row stride (floats): 36*l mod 64 -> conflict-free
#define NCH   (512 / DC)   // 16 chunks

#define HAS_ASYNC_LDS __has_builtin(__builtin_amdgcn_global_load_async_to_lds_b128)

#if HAS_ASYNC_LDS
typedef v4i __attribute__((address_space(1)))* g16_t;   // global 16B block ptr
typedef v4i __attribute__((address_space(3)))* l16_t;   // LDS 16B block ptr
#endif

// ---- 16-byte global -> LDS copy (async on CDNA5, sync fallback) -----------
__device__ __forceinline__ void copy16_g2l(const float* g, float* l) {
#if HAS_ASYNC_LDS
    __builtin_amdgcn_global_load_async_to_lds_b128(
        (g16_t)(void*)(float*)g, (l16_t)(void*)l, /*offset=*/0, /*cpol=*/0);
#else
    *(float4*)l = *(const float4*)g;
#endif
}

template <int PENDING>
__device__ __forceinline__ void wait_g2l() {
#if HAS_ASYNC_LDS
#if __has_builtin(__builtin_amdgcn_s_wait_asynccnt)
    __builtin_amdgcn_s_wait_asynccnt((short)PENDING);
#else
    asm volatile("s_wait_asynccnt %0" :: "i"(PENDING));
#endif
#endif
}

// Stage one D-chunk: x 128x32 (4 b128/thread) + embed 64x32 (2 b128/thread)
// => 6 async ops in flight per thread per stage.
__device__ __forceinline__ void stage_chunk(const float* __restrict__ x,
                                            const float* __restrict__ embed,
                                            float* xsbuf, float* esbuf,
                                            int tid, int m0, int n0, int kc, int D) {
    #pragma unroll
    for (int i = 0; i < 4; ++i) {
        int idx = tid + i * 256;          // 0..1023
        int row = idx >> 3;               // 0..127
        int seg = (idx & 7) << 2;         // 0,4,...,28
        copy16_g2l(x + (size_t)(m0 + row) * D + kc + seg, xsbuf + row * LDSS + seg);
    }
    #pragma unroll
    for (int i = 0; i < 2; ++i) {
        int idx = tid + i * 256;          // 0..511
        int row = idx >> 3;               // 0..63
        int seg = (idx & 7) << 2;
        copy16_g2l(embed + (size_t)(n0 + row) * D + kc + seg, esbuf + row * LDSS + seg);
    }
}

// ---------------------------------------------------------------------------
// Kernel 1: row-wise squared L2 norm (block per row)
// ---------------------------------------------------------------------------
__global__ __launch_bounds__(256)
void vq_rowsq_kernel(const float* __restrict__ v, float* __restrict__ out, int D) {
    __shared__ float red[256];
    const int row = blockIdx.x;
    const float* p = v + (size_t)row * D;
    float s = 0.0f;
    for (int d = threadIdx.x; d < D; d += 256) {
        float t = p[d];
        s += t * t;
    }
    red[threadIdx.x] = s;
    __syncthreads();
    for (int o = 128; o > 0; o >>= 1) {
        if (threadIdx.x < o) red[threadIdx.x] += red[threadIdx.x + o];
        __syncthreads();
    }
    if (threadIdx.x == 0) out[row] = red[0];
}

// ---------------------------------------------------------------------------
// Kernel 2: fused distance GEMM (fp32 WMMA) + per-row argmax scatter.
// Block: 256 threads = 8 waves laid out 4(M) x 2(N); each wave owns 32x32
// (2x2 accumulators). Double-buffered async LDS staging over D.
// ---------------------------------------------------------------------------
__global__ __launch_bounds__(256)
void vq_dist_wmma_kernel(const float* __restrict__ x,
                         const float* __restrict__ embed,
                         const float* __restrict__ xsq,
                         const float* __restrict__ esq,
                         float* __restrict__ dist,
                         unsigned long long* __restrict__ packed,
                         int K, int D) {
    __shared__ float xs[2][TM * LDSS];   // 2 x 128x32 chunk of x  (padded)
    __shared__ float es[2][TN * LDSS];   // 2 x  64x32 chunk of E  (padded)

    const int tid  = threadIdx.x;
    const int lane = tid & 31;
    const int wave = tid >> 5;            // 0..7
    const int wm   = wave >> 1;           // 0..3 -> 32-row band
    const int wn   = wave & 1;            // 0..1 -> 32-col band
    const int half = lane >> 4;           // 0/1
    const int l16  = lane & 15;

    const int m0 = blockIdx.y * TM;
    const int n0 = blockIdx.x * TN;

    v8f acc00 = {}, acc01 = {}, acc10 = {}, acc11 = {};

    stage_chunk(x, embed, xs[0], es[0], tid, m0, n0, 0, D);

    int buf = 0;
    for (int c = 0; c < NCH; ++c) {
        if (c + 1 < NCH) {
            stage_chunk(x, embed, xs[buf ^ 1], es[buf ^ 1], tid, m0, n0, (c + 1) * DC, D);
            wait_g2l<6>();    // previous stage (6 older ops) has landed
        } else {
            wait_g2l<0>();
        }
        __syncthreads();

        // A frags: rows wm*32 + {0..15}, {16..31}; B frags: cols wn*32 + ...
        // 16x4 f32 layout: lane l<16 -> [row=l][k,k+1]; l>=16 -> [row=l-16][k+2,k+3]
        const float* ax = &xs[buf][(wm * 32 + l16) * LDSS + 2 * half];
        const float* bx = &es[buf][(wn * 32 + l16) * LDSS + 2 * half];

        #pragma unroll
        for (int k = 0; k < DC; k += 4) {
            v2f a0 = *(const v2f*)(ax + k);
            v2f a1 = *(const v2f*)(ax + 16 * LDSS + k);
            v2f b0 = *(const v2f*)(bx + k);
            v2f b1 = *(const v2f*)(bx + 16 * LDSS + k);
            acc00 = __builtin_amdgcn_wmma_f32_16x16x4_f32(false, a0, false, b0, (short)0, acc00, false, false);
            acc01 = __builtin_amdgcn_wmma_f32_16x16x4_f32(false, a0, false, b1, (short)0, acc01, false, false);
            acc10 = __builtin_amdgcn_wmma_f32_16x16x4_f32(false, a1, false, b0, (short)0, acc10, false, false);
            acc11 = __builtin_amdgcn_wmma_f32_16x16x4_f32(false, a1, false, b1, (short)0, acc11, false, false);
        }
        __syncthreads();
        buf ^= 1;
    }

    // ---- epilogue: dist = 2*dot - |x|^2 - |e|^2 ; fused row-argmax --------
    // C/D layout: VGPR r, lanes 0-15 -> M=r; lanes 16-31 -> M=r+8; N = lane%16
    const int mBase = m0 + wm * 32;
    const int nBase = n0 + wn * 32;
    const float eq0 = esq[nBase + l16];
    const float eq1 = esq[nBase + 16 + l16];

    #pragma unroll
    for (int mi = 0; mi < 2; ++mi) {
        v8f aN = mi ? acc10 : acc00;   // cols nBase + l16
        v8f aO = mi ? acc11 : acc01;   // cols nBase + 16 + l16
        #pragma unroll
        for (int r = 0; r < 8; ++r) {
            const int   m  = mBase + mi * 16 + r + half * 8;
            const float xq = xsq[m];
            const float d0 = 2.0f * aN[r] - xq - eq0;
            const float d1 = 2.0f * aO[r] - xq - eq1;
            dist[(size_t)m * K + nBase + l16]      = d0;
            dist[(size_t)m * K + nBase + 16 + l16] = d1;

            // local best of this lane's two columns (tie -> smaller col = d0)
            float bv = d0; int bc = nBase + l16;
            if (d1 > bv) { bv = d1; bc = nBase + 16 + l16; }
            // reduce across the 16 lanes of this half (masks stay within half)
            #pragma unroll
            for (int off = 8; off >= 1; off >>= 1) {
                float ov = __shfl_xor(bv, off, 32);
                int   oc = __shfl_xor(bc, off, 32);
                if (ov > bv || (ov == bv && oc < bc)) { bv = ov; bc = oc; }
            }
            if (l16 == 0) {
                unsigned int ub = __float_as_uint(bv);
                unsigned int u  = (ub & 0x80000000u) ? ~ub : (ub | 0x80000000u);
                unsigned long long pk =
                    ((unsigned long long)u << 32) | (unsigned int)(0x7FFFFFFF - bc);
                atomicMax(&packed[m], pk);
            }
        }
    }
}

// ---------------------------------------------------------------------------
// Kernel 3: decode packed argmax -> int index + float output
// ---------------------------------------------------------------------------
__global__ __launch_bounds__(256)
void vq_decode_kernel(const unsigned long long* __restrict__ packed,
                      int* __restrict__ ind, float* __restrict__ ind_f, int N) {
    int i = blockIdx.x * 256 + threadIdx.x;
    if (i < N) {
        unsigned long long p = packed[i];
        int k = 0x7FFFFFFF - (int)(unsigned int)(p & 0xFFFFFFFFull);
        ind[i]   = k;
        ind_f[i] = (float)k;
    }
}

// ---------------------------------------------------------------------------
// Kernel 4: zero workspace accumulators (bins, total, embed_sum, packed)
// ---------------------------------------------------------------------------
__global__ __launch_bounds__(256)
void vq_zero_kernel(float* __restrict__ p, size_t n) {
    size_t i = (size_t)blockIdx.x * 256 + threadIdx.x;
    if (i < n) p[i] = 0.0f;
}

// ---------------------------------------------------------------------------
// Kernel 5: gather quantize rows + scatter segment sums (block per row)
// ---------------------------------------------------------------------------
__global__ __launch_bounds__(256)
void vq_gather_scatter_kernel(const float* __restrict__ x,
                              const float* __restrict__ embed,
                              const int* __restrict__ ind,
                              float* __restrict__ quant,
                              float* __restrict__ bins,
                              float* __restrict__ esum, int D) {
    const int row = blockIdx.x;
    const int k   = ind[row];
    const float* e  = embed + (size_t)k * D;
    const float* xr = x + (size_t)row * D;
    float* q  = quant + (size_t)row * D;
    float* es = esum + (size_t)k * D;
    for (int d = threadIdx.x; d < D; d += 256) {
        q[d] = e[d];
        atomicAdd(&es[d], xr[d]);
    }
    if (threadIdx.x == 0) atomicAdd(&bins[k], 1.0f);
}

// ---------------------------------------------------------------------------
// Kernel 6: new_cluster_size EMA + total reduction
// ---------------------------------------------------------------------------
__global__ __launch_bounds__(256)
void vq_ema_cs_kernel(const float* __restrict__ cs, const float* __restrict__ bins,
                      float* __restrict__ ncs, float* __restrict__ total, int K) {
    int k = blockIdx.x * 256 + threadIdx.x;
    if (k < K) {
        float v = cs[k] * DECAY + (1.0f - DECAY) * bins[k];
        ncs[k] = v;
        atomicAdd(total, v);
    }
}

// ---------------------------------------------------------------------------
// Kernel 7: new_embed_avg EMA + Laplace-smoothed new_embed (block per k)
// ---------------------------------------------------------------------------
__global__ __launch_bounds__(256)
void vq_ema_embed_kernel(const float* __restrict__ ea, const float* __restrict__ esum,
                         const float* __restrict__ ncs, const float* __restrict__ total,
                         float* __restrict__ nea, float* __restrict__ ne,
                         int K, int D) {
    const int   k   = blockIdx.x;
    const float t   = *total;
    const float sm  = (ncs[k] + EPSILON) / (t + (float)K * EPSILON) * t;
    const float inv = 1.0f / sm;
    const size_t base = (size_t)k * D;
    for (int d = threadIdx.x; d < D; d += 256) {
        float v = ea[base + d] * DECAY + (1.0f - DECAY) * esum[base + d];
        nea[base + d] = v;
        ne[base + d]  = v * inv;
    }
}

// ---------------------------------------------------------------------------
// Host-side launch
// ---------------------------------------------------------------------------
extern "C" void kernel_launch(void* const* d_in, const int* in_sizes, int n_in,
                              void* d_out, int out_size, void* d_ws, size_t ws_size,
                              hipStream_t stream) {
    const int B = 16, S = 2048, D = 512, K = 2048;
    const int N = B * S;                       // 32768

    const float* x     = (const float*)d_in[0];   // [N, D]
    const float* embed = (const float*)d_in[1];   // [K, D]
    const float* cs    = (const float*)d_in[2];   // [K]
    const float* ea    = (const float*)d_in[3];   // [K, D]

    // ---- output layout (flat float32, return order) ----
    float* out = (float*)d_out;
    float* o_quant = out;                                        // N*D
    float* o_ind   = o_quant + (size_t)N * D;                    // N
    float* o_dist  = o_ind + N;                                  // N*K
    float* o_ncs   = o_dist + (size_t)N * K;                     // K
    float* o_nea   = o_ncs + K;                                  // K*D
    float* o_ne    = o_nea + (size_t)K * D;                      // K*D

    // ---- workspace layout (floats; bins..packed kept contiguous for zeroing)
    float* ws    = (float*)d_ws;
    float* xsq   = ws;                               // N
    float* esq   = xsq + N;                          // K
    float* bins  = esq + K;                          // K
    float* total = bins + K;                         // 1 (+3 pad)
    float* esum  = total + 4;                        // K*D
    unsigned long long* packed =
        (unsigned long long*)(esum + (size_t)K * D);  // N u64 (8B aligned)
    int*   ind   = (int*)(packed + N);               // N ints

    // 1) squared norms
    vq_rowsq_kernel<<<N, 256, 0, stream>>>(x, xsq, D);
    vq_rowsq_kernel<<<K, 256, 0, stream>>>(embed, esq, D);

    // 2) zero atomic accumulators: bins + total pad + esum + packed
    {
        size_t nz = (size_t)K + 4 + (size_t)K * D + 2 * (size_t)N;
        vq_zero_kernel<<<(unsigned)((nz + 255) / 256), 256, 0, stream>>>(bins, nz);
    }

    // 3) fused distance GEMM + argmax scatter (WMMA fp32, async LDS staging)
    {
        dim3 grid(K / TN, N / TM);   // (32, 256)
        vq_dist_wmma_kernel<<<grid, 256, 0, stream>>>(x, embed, xsq, esq,
                                                      o_dist, packed, K, D);
    }

    // 4) decode argmax
    vq_decode_kernel<<<(N + 255) / 256, 256, 0, stream>>>(packed, ind, o_ind, N);

    // 5) gather quantize + segment sums
    vq_gather_scatter_kernel<<<N, 256, 0, stream>>>(x, embed, ind, o_quant, bins, esum, D);

    // 6) cluster-size EMA + total
    vq_ema_cs_kernel<<<(K + 255) / 256, 256, 0, stream>>>(cs, bins, o_ncs, total, K);

    // 7) embed-avg EMA + smoothed embed
    vq_ema_embed_kernel<<<K, 256, 0, stream>>>(ea, esum, o_ncs, total, o_nea, o_ne, K, D);
}